// BackgroundSubtractorModule_19155554140806
// MI455X (gfx1250) — compile-verified
//
#include <hip/hip_runtime.h>
#include <hip/hip_bf16.h>

// ---- problem constants (match reference shapes exactly) ----
#define T_FRAMES 450
#define H_DIM    384
#define W_DIM    384
#define HW       (H_DIM * W_DIM)       // 147456
#define WIN      15
#define NWIN     (T_FRAMES / WIN)      // 30 full windows, no pad (450 = 30*15)
#define EPSV     1e-5f
#define NTHRESH  1e-6f

typedef __attribute__((ext_vector_type(2))) float v2f;
typedef __attribute__((ext_vector_type(8))) float v8f;
typedef __attribute__((ext_vector_type(4))) float f4;

// monotonic float<->uint mapping (total order, handles negatives)
__device__ __forceinline__ unsigned fmono(float f) {
    unsigned u = __float_as_uint(f);
    return (u & 0x80000000u) ? ~u : (u | 0x80000000u);
}
__device__ __forceinline__ float fmono_inv(unsigned u) {
    unsigned b = (u & 0x80000000u) ? (u & 0x7FFFFFFFu) : ~u;
    return __uint_as_float(b);
}

__device__ __forceinline__ unsigned umin_(unsigned a, unsigned b) { return a < b ? a : b; }
__device__ __forceinline__ unsigned umax_(unsigned a, unsigned b) { return a > b ? a : b; }

// ws layout (uints): [0] = fmono(global max of raw video)
//                    [1 .. 450]    = per-frame min of bgsub (raw f32 bits, nonneg)
//                    [451 .. 900]  = per-frame max of bgsub (raw f32 bits, nonneg)
__global__ void init_ws_kernel(unsigned* __restrict__ ws) {
    int i = threadIdx.x;
    if (i == 0) ws[0] = 0u;
    if (i >= 1 && i <= T_FRAMES) ws[i] = 0x7F7FFFFFu;               // +FLT_MAX bits
    if (i >= T_FRAMES + 1 && i <= 2 * T_FRAMES) ws[i] = 0u;         // 0.0f bits
}

// Pass A: RGB -> luma (write to d_out as gray), plus global max of raw input.
__global__ __launch_bounds__(256)
void gray_max_kernel(const float* __restrict__ vid, float* __restrict__ gray,
                     unsigned* __restrict__ ws, int nvec) {
    int stride = gridDim.x * blockDim.x;
    float lmax = -3.4e38f;
    for (int i = blockIdx.x * blockDim.x + threadIdx.x; i < nvec; i += stride) {
        const f4* vp = (const f4*)(vid + (size_t)i * 12);   // 4 pixels * RGB
        f4 a = vp[0], b = vp[1], c = vp[2];
        f4 g;
        g.x = 0.299f * a.x + 0.587f * a.y + 0.114f * a.z;
        g.y = 0.299f * a.w + 0.587f * b.x + 0.114f * b.y;
        g.z = 0.299f * b.z + 0.587f * b.w + 0.114f * c.x;
        g.w = 0.299f * c.y + 0.587f * c.z + 0.114f * c.w;
        ((f4*)gray)[i] = g;
        lmax = fmaxf(lmax, fmaxf(fmaxf(a.x, a.y), fmaxf(a.z, a.w)));
        lmax = fmaxf(lmax, fmaxf(fmaxf(b.x, b.y), fmaxf(b.z, b.w)));
        lmax = fmaxf(lmax, fmaxf(fmaxf(c.x, c.y), fmaxf(c.z, c.w)));
    }
    unsigned m = fmono(lmax);
    #pragma unroll
    for (int off = 1; off < 32; off <<= 1) m = umax_(m, __shfl_xor(m, off, 32));
    __shared__ unsigned sm[8];
    int wid = threadIdx.x >> 5;
    if ((threadIdx.x & 31) == 0) sm[wid] = m;
    __syncthreads();
    if (threadIdx.x == 0) {
        unsigned bm = sm[0];
        for (int wv = 1; wv < (int)(blockDim.x >> 5); ++wv) bm = umax_(bm, sm[wv]);
        atomicMax(&ws[0], bm);
    }
}

// Pass B: in-place gray -> bgsub per 15-frame window, plus per-frame min/max.
// Each wave handles 16 pixels: lane n and lane n+16 own pixel (base+n).
// Temporal sums done with V_WMMA_F32_16X16X4_F32:
//   A = ones(16x4)  (VGPR0/1 = 1.0 in all lanes)
//   B = 4 frames x 16 pixels (column n = pixel n; lane pair (n, n+16) holds 4 K-slots)
//   D column sums land in every lane's d[0] for its own pixel -> exact f32 reduction
// (with A==1 the sum is invariant to the exact K-slot<->lane mapping of B).
__global__ __launch_bounds__(256)
void bgsub_kernel(float* __restrict__ data, unsigned* __restrict__ ws) {
    __shared__ unsigned smin[WIN], smax[WIN];
    int tid = threadIdx.x;
    if (tid < WIN) { smin[tid] = 0x7F7FFFFFu; smax[tid] = 0u; }
    __syncthreads();

    float vmax = fmono_inv(ws[0]);
    float scale = (vmax > 1.0f) ? (1.0f / 255.0f) : 1.0f;

    int w    = blockIdx.y;            // window index
    int lane = tid & 31;
    int wave = tid >> 5;
    int half = lane >> 4;             // 0: frames {4j,4j+1}, 1: frames {4j+2,4j+3}
    int n    = lane & 15;             // pixel within the wave's 16-pixel group

    v2f ones; ones.x = 1.0f; ones.y = 1.0f;

    for (int c = 0; c < 4; ++c) {     // 4 pixel-chunks per block -> 512 pixels/block
        int p = blockIdx.x * 512 + c * 128 + wave * 16 + n;

        float g[4][2];
        v8f acc = {};
        #pragma unroll
        for (int j = 0; j < 4; ++j) {
            #pragma unroll
            for (int s = 0; s < 2; ++s) {
                int f  = 4 * j + 2 * half + s;
                int fc = f < WIN ? f : (WIN - 1);              // clamp dummy slot (f==15)
                float val = data[(size_t)(w * WIN + fc) * HW + p] * scale;
                g[j][s] = (f < WIN) ? val : 0.0f;              // dummy contributes 0
            }
            v2f b; b.x = g[j][0]; b.y = g[j][1];
            acc = __builtin_amdgcn_wmma_f32_16x16x4_f32(
                false, ones, false, b, (short)0, acc, false, false);
        }
        float mean = acc[0] * (1.0f / WIN);

        v8f acc2 = {};
        #pragma unroll
        for (int j = 0; j < 4; ++j) {
            v2f b;
            #pragma unroll
            for (int s = 0; s < 2; ++s) {
                int f = 4 * j + 2 * half + s;
                float d = g[j][s] - mean;
                float q = (f < WIN) ? d * d : 0.0f;
                if (s == 0) b.x = q; else b.y = q;
            }
            acc2 = __builtin_amdgcn_wmma_f32_16x16x4_f32(
                false, ones, false, b, (short)0, acc2, false, false);
        }
        float sigma = sqrtf(acc2[0] * (1.0f / (WIN - 1))) + EPSV;
        float inv   = 1.0f / sigma;

        #pragma unroll
        for (int j = 0; j < 4; ++j) {
            #pragma unroll
            for (int s = 0; s < 2; ++s) {
                int f = 4 * j + 2 * half + s;
                if (f < WIN) {                                  // uniform per 16-lane half
                    float bg = fabsf(g[j][s] - mean) * inv;
                    data[(size_t)(w * WIN + f) * HW + p] = bg;
                    // half-wave (16-lane) min/max reduction; bgsub >= 0 so raw
                    // f32 bits order as uint. xor masks <16 stay within halves.
                    unsigned mn = __float_as_uint(bg);
                    unsigned mx = mn;
                    #pragma unroll
                    for (int o = 1; o < 16; o <<= 1) {
                        mn = umin_(mn, __shfl_xor(mn, o, 32));
                        mx = umax_(mx, __shfl_xor(mx, o, 32));
                    }
                    if (n == 0) {
                        atomicMin(&smin[f], mn);
                        atomicMax(&smax[f], mx);
                    }
                }
            }
        }
    }
    __syncthreads();
    if (tid < WIN) {
        atomicMin(&ws[1 + w * WIN + tid], smin[tid]);
        atomicMax(&ws[1 + T_FRAMES + w * WIN + tid], smax[tid]);
    }
}

// Pass C: in-place per-frame min-max normalize (float4 vectorized; HW % 4 == 0
// so a float4 never crosses a frame boundary).
__global__ __launch_bounds__(256)
void normalize_kernel(float* __restrict__ data, const unsigned* __restrict__ ws, int nvec) {
    int i = blockIdx.x * blockDim.x + threadIdx.x;
    if (i >= nvec) return;
    int t = (i * 4) / HW;
    float mn = __uint_as_float(ws[1 + t]);
    float mx = __uint_as_float(ws[1 + T_FRAMES + t]);
    float rng = mx - mn;
    f4 v = ((const f4*)data)[i];
    if (rng > NTHRESH) {
        float inv = 1.0f / rng;
        v.x = (v.x - mn) * inv;
        v.y = (v.y - mn) * inv;
        v.z = (v.z - mn) * inv;
        v.w = (v.w - mn) * inv;
    }
    ((f4*)data)[i] = v;
}

extern "C" void kernel_launch(void* const* d_in, const int* in_sizes, int n_in,
                              void* d_out, int out_size, void* d_ws, size_t ws_size,
                              hipStream_t stream) {
    (void)in_sizes; (void)n_in; (void)out_size; (void)ws_size;
    const float* vid = (const float*)d_in[0];
    float*       out = (float*)d_out;              // gray (pass A) -> bgsub -> normalized
    unsigned*    ws  = (unsigned*)d_ws;            // 901 uints used

    init_ws_kernel<<<1, 1024, 0, stream>>>(ws);

    int nvecA = T_FRAMES * HW / 4;                 // 16,588,800 float4 groups
    gray_max_kernel<<<2048, 256, 0, stream>>>(vid, out, ws, nvecA);

    dim3 gB(HW / 512, NWIN);                       // (288, 30)
    bgsub_kernel<<<gB, 256, 0, stream>>>(out, ws);

    int nvecC = T_FRAMES * HW / 4;
    normalize_kernel<<<(nvecC + 255) / 256, 256, 0, stream>>>(out, ws, nvecC);
}